// LivingLooper_39213051412544
// MI455X (gfx1250) — compile-verified
//
#include <hip/hip_runtime.h>
#include <math.h>

// ---- problem constants (match reference) ----
#define NM     1067            // N_MEMORY
#define L      8               // N_LOOPS
#define D      128             // N_LATENT
#define CTX    64              // N_CONTEXT
#define F      (CTX * L * D)   // 65536 flat feature count
#define BLOCK  2048
#define NT     (D / 16)        // 8 output tiles of 16
#define KB     128             // K-blocks per (l, tile)
#define KS     (F / KB)        // 512 K per block

typedef __attribute__((ext_vector_type(2))) float v2f;
typedef __attribute__((ext_vector_type(8))) float v8f;

// ---------------------------------------------------------------------------
// K1: gather ring-buffer window (with active-loop z_in overrides) and compute
//     x[l,f] = tanh((flat[f] - center[l,f]) * 0.5)
// ---------------------------------------------------------------------------
__global__ void build_x(const float* __restrict__ memory,
                        const float* __restrict__ z_in,
                        const float* __restrict__ center,
                        const int* __restrict__ ri_p,
                        const int* __restrict__ al_p,
                        const int* __restrict__ lc_p,
                        float* __restrict__ xfeat) {
  int idx = blockIdx.x * blockDim.x + threadIdx.x;
  if (idx >= L * F) return;
  int l  = idx / F;
  int f  = idx - l * F;
  int c  = f / (L * D);           // context position 0..CTX-1
  int r  = f - c * (L * D);
  int lp = r / D;                 // loop within window
  int d  = r - lp * D;            // latent dim

  int ri   = ri_p[0];
  int iact = al_p[0] - 1;
  int lc   = lc_p[0];

  int row = (ri - CTX + c) % NM;  if (row < 0) row += NM;
  int dt  = (ri - row) % NM;      if (dt  < 0) dt  += NM;

  float val;
  if (lp == iact && dt <= lc) {
    val = z_in[d];                            // freshly recorded latent
  } else {
    val = memory[((size_t)row * L + lp) * D + d];
  }
  xfeat[idx] = tanhf((val - center[idx]) * 0.5f);
}

// ---------------------------------------------------------------------------
// K2: WMMA f32 16x16x4 partial matvec. One wave per block (scalar control),
//     per-lane base pointers hoisted out of the loop so every unrolled load
//     is VGPR-base + constant instruction-offset.
//     A = x replicated across all 16 M rows; B = W[l, k..k+3, n0..n0+15].
//     Row 0 of C (lanes 0..15, VGPR0) holds the partial z for 16 outputs.
// ---------------------------------------------------------------------------
__global__ void __launch_bounds__(32)
gemm_partial(const float* __restrict__ W,
             const float* __restrict__ xfeat,
             float* __restrict__ partials) {
  const int wid  = blockIdx.x;            // scalar: (l*NT + ntile)*KB + kb
  const int lane = threadIdx.x & 31;

  const int l     = wid / (NT * KB);
  const int rem   = wid - l * (NT * KB);
  const int ntile = rem / KB;
  const int kb    = rem - ntile * KB;

  const int n  = lane & 15;
  const int kr = (lane >> 4) << 1;        // 0 for lanes 0-15, 2 for lanes 16-31

  // Per-lane base pointers: all lane-varying address math happens ONCE here.
  // Inside the loop only the scalar offset k*D / k changes.
  const float* __restrict__ Wp =
      W + (size_t)l * F * D + (size_t)ntile * 16 + (size_t)(kr * D + n);
  const float* __restrict__ xp = xfeat + (size_t)l * F + kr;

  v8f acc = {};
  const int k0 = kb * KS;
#pragma unroll 8
  for (int k = k0; k < k0 + KS; k += 4) {
    const size_t ko = (size_t)k * D;      // scalar offset
    v2f a, b;
    a.x = xp[k];
    a.y = xp[k + 1];
    b.x = Wp[ko];
    b.y = Wp[ko + D];
    // (neg_a, A, neg_b, B, c_mod, C, reuse_a, reuse_b)
    acc = __builtin_amdgcn_wmma_f32_16x16x4_f32(false, a, false, b,
                                                (short)0, acc, false, false);
  }
  if (lane < 16) {
    partials[(size_t)wid * 16 + lane] = acc[0];   // M=0 row
  }
}

// ---------------------------------------------------------------------------
// K3: deterministic reduction over k-blocks + bias + target_process_inv +
//     clamp + active-loop override; writes the zs output block.
// ---------------------------------------------------------------------------
__global__ void reduce_z(const float* __restrict__ partials,
                         const float* __restrict__ bias,
                         const float* __restrict__ z_min,
                         const float* __restrict__ z_max,
                         const float* __restrict__ margin,
                         const float* __restrict__ z_in,
                         const int* __restrict__ al_p,
                         float* __restrict__ out_zs) {
  int idx = blockIdx.x * blockDim.x + threadIdx.x;   // l*D + d
  if (idx >= L * D) return;
  int l = idx / D;
  int d = idx - l * D;
  int ntile = d >> 4;
  int nn    = d & 15;

  const float* p = partials + (size_t)((l * NT + ntile) * KB) * 16 + nn;
  float s = 0.0f;
  for (int kb = 0; kb < KB; ++kb) s += p[(size_t)kb * 16];

  float z = s + bias[idx];
  if (z > 1.0f)       z = 2.0f * sqrtf(z) - 1.0f;
  else if (z < -1.0f) z = 1.0f - 2.0f * sqrtf(-z);

  float lo = z_min[idx] - margin[d];
  float hi = z_max[idx] + margin[d];
  z = fminf(fmaxf(z, lo), hi);

  int iact = al_p[0] - 1;
  if (l == iact) z = z_in[d];        // mem1[record_index, i_act] == z_in
  out_zs[idx] = z;
}

// ---------------------------------------------------------------------------
// K4: materialize mem2 = memory with ring-buffer updates + scatter of zs row.
// ---------------------------------------------------------------------------
__global__ void write_mem2(const float* __restrict__ memory,
                           const float* __restrict__ z_in,
                           const float* __restrict__ zs,
                           const int* __restrict__ ri_p,
                           const int* __restrict__ al_p,
                           const int* __restrict__ lc_p,
                           float* __restrict__ out_mem) {
  int idx = blockIdx.x * blockDim.x + threadIdx.x;
  if (idx >= NM * L * D) return;
  int r   = idx / (L * D);
  int rem = idx - r * (L * D);
  int l   = rem / D;
  int d   = rem - l * D;

  int ri = ri_p[0];
  float val;
  if (r == ri) {
    val = zs[l * D + d];             // row record_index == zs for all loops
  } else {
    int dt = (ri - r) % NM; if (dt < 0) dt += NM;
    int iact = al_p[0] - 1;
    int lc   = lc_p[0];
    if (l == iact && dt <= lc) val = z_in[d];
    else                       val = memory[idx];
  }
  out_mem[idx] = val;
}

// ---------------------------------------------------------------------------
// K5: masked crossfade on decoded audio.
// ---------------------------------------------------------------------------
__global__ void audio_fade(const float* __restrict__ y,
                           const float* __restrict__ mask,
                           float* __restrict__ out_y) {
  int idx = blockIdx.x * blockDim.x + threadIdx.x;
  if (idx >= L * BLOCK) return;
  int l = idx / BLOCK;
  int t = idx - l * BLOCK;
  float fade = (float)t / (float)(BLOCK - 1);
  float m = mask[L + l] * fade + mask[l] * (1.0f - fade);
  out_y[idx] = y[idx] * m;
}

// ---------------------------------------------------------------------------
extern "C" void kernel_launch(void* const* d_in, const int* in_sizes, int n_in,
                              void* d_out, int out_size, void* d_ws, size_t ws_size,
                              hipStream_t stream) {
  const float* memory  = (const float*)d_in[0];
  const float* z_in    = (const float*)d_in[1];
  const float* weights = (const float*)d_in[2];
  const float* center  = (const float*)d_in[3];
  const float* bias    = (const float*)d_in[4];
  const float* z_min   = (const float*)d_in[5];
  const float* z_max   = (const float*)d_in[6];
  const float* margin  = (const float*)d_in[7];
  const float* mask    = (const float*)d_in[8];
  const float* y_audio = (const float*)d_in[9];
  const int*   ri      = (const int*)d_in[10];
  const int*   al      = (const int*)d_in[11];
  const int*   lc      = (const int*)d_in[12];
  // d_in[13] = context (fixed at 64 by shapes)

  float* out     = (float*)d_out;
  float* out_y   = out;                    // [L*BLOCK]   = 16384
  float* out_zs  = out + L * BLOCK;        // [L*D]       = 1024
  float* out_mem = out_zs + L * D;         // [NM*L*D]    = 1092608

  float* xfeat    = (float*)d_ws;                  // L*F        floats (2 MB)
  float* partials = xfeat + (size_t)L * F;         // L*NT*KB*16 floats (0.5 MB)

  build_x<<<(L * F + 255) / 256, 256, 0, stream>>>(
      memory, z_in, center, ri, al, lc, xfeat);

  int nwaves = L * NT * KB;                        // 8192 waves
  gemm_partial<<<nwaves, 32, 0, stream>>>(         // one wave per block
      weights, xfeat, partials);

  reduce_z<<<(L * D + 255) / 256, 256, 0, stream>>>(
      partials, bias, z_min, z_max, margin, z_in, al, out_zs);

  write_mem2<<<(NM * L * D + 255) / 256, 256, 0, stream>>>(
      memory, z_in, out_zs, ri, al, lc, out_mem);

  audio_fade<<<(L * BLOCK + 255) / 256, 256, 0, stream>>>(
      y_audio, mask, out_y);
}